// FLUX_20529943675547
// MI455X (gfx1250) — compile-verified
//
#include <hip/hip_runtime.h>
#include <hip/hip_bf16.h>

typedef float v2f __attribute__((ext_vector_type(2)));
typedef float v4f __attribute__((ext_vector_type(4)));
typedef float v8f __attribute__((ext_vector_type(8)));

#define NB    16384
#define NMOD  256
#define NGEN  50
#define NHID  8
#define NCMP  70
#define XCOLS (NMOD * NGEN)      // 12800
#define LDSS  260                // floats: 260%64=4 -> lanes hit distinct banks; 260*4%16==0 -> rows 16B aligned

// Branch-free tanh on hardware transcendentals: tanh(x) = 1 - 2/(exp2(2*log2e*x)+1)
// Correct saturation at +/-inf (exp2 overflow -> rcp(inf)=0 -> 1; underflow -> -1).
__device__ __forceinline__ float tanh_fast(float x) {
    const float e = __builtin_amdgcn_exp2f(x * 2.8853900817779268f); // 2*log2(e)
    return 1.0f - 2.0f * __builtin_amdgcn_rcpf(e + 1.0f);
}
__device__ __forceinline__ float tanhshrink_f(float v) {
    return v - tanh_fast(v);
}

__global__ __launch_bounds__(32)
void flux_fused_kernel(const float* __restrict__ x,
                       const float* __restrict__ W1,
                       const float* __restrict__ W2,
                       const float* __restrict__ b2,
                       const float* __restrict__ cm,
                       float* __restrict__ out_m,
                       float* __restrict__ out_c)
{
    __shared__ float mt[16 * LDSS];          // m_tile[16 cells][256 modules]

    const int lane = threadIdx.x;            // 0..31 (wave32)
    const int half = lane >> 4;              // 0 or 1
    const int l15  = lane & 15;
    const int b0   = blockIdx.x * 16;        // first cell of this tile

    // Lane's A-matrix row (M = cell b0+l15), f32 16x4 A layout: K = 4*kc + 2*half + r
    const float* __restrict__ xrow = x + (size_t)(b0 + l15) * XCOLS;

    // ---------------- per-module MLP: h = ts(x W1), m = ts(h W2 + b2) ----------------
    for (int mod = 0; mod < NMOD; ++mod) {
        __builtin_prefetch(xrow + (mod + 1) * NGEN, 0, 1);   // global_prefetch_b8

        // ---- B fragments of W1[mod] (50x8 -> K=52, N=16 zero-padded) ----
        // lane holds N=l15, K=4kc+2half+r; single base + constant offsets
        const float* __restrict__ w1m = W1 + mod * (NGEN * NHID);
        const float* __restrict__ q   = w1m + 16 * half + l15;
        const bool hv = (l15 < NHID);
        v2f bfrag[13];
        #pragma unroll
        for (int kc = 0; kc < 12; ++kc) {
            bfrag[kc][0] = hv ? q[kc * 32]     : 0.f;
            bfrag[kc][1] = hv ? q[kc * 32 + 8] : 0.f;
        }
        {   // chunk 12: K = 48+2half; rows 50,51 (half==1) are zero padding
            const bool lv = hv && (half == 0);
            bfrag[12][0] = lv ? q[12 * 32]     : 0.f;
            bfrag[12][1] = lv ? q[12 * 32 + 8] : 0.f;
        }

        // ---- A fragments: constant-offset b64 loads from two bases ----
        const float* __restrict__ p = xrow + mod * NGEN + 2 * half;
        v8f acc = {};
        #pragma unroll
        for (int kc = 0; kc < 12; ++kc) {
            const v2f a = *(const v2f*)(p + kc * 4);             // imm-offset global_load_b64
            acc = __builtin_amdgcn_wmma_f32_16x16x4_f32(
                      false, a, false, bfrag[kc], (short)0, acc, false, false);
        }
        {   // chunk 12: both halves load cols 48,49 (K=50,51 multiply zero B rows)
            const v2f a = *(const v2f*)(xrow + mod * NGEN + 48);
            acc = __builtin_amdgcn_wmma_f32_16x16x4_f32(
                      false, a, false, bfrag[12], (short)0, acc, false, false);
        }

        // ---- tanhshrink + layer-2 dot (reduce h across lanes of each 16-half) ----
        const float w2v = hv ? W2[mod * NHID + l15] : 0.f;
        const float b2v = b2[mod];

        float mv[8];
        #pragma unroll
        for (int r = 0; r < 8; ++r) {
            float s = tanhshrink_f(acc[r]) * w2v;
            s += __shfl_xor(s, 1, 32);
            s += __shfl_xor(s, 2, 32);
            s += __shfl_xor(s, 4, 32);     // h-cols 8..15 are zero: 3 levels suffice
            mv[r] = tanhshrink_f(s + b2v);
        }

        if (l15 == 0) {                     // lanes 0 & 16 commit their half's 8 cells
            #pragma unroll
            for (int r = 0; r < 8; ++r)
                mt[(r + 8 * half) * LDSS + mod] = mv[r];
        }
    }
    __syncthreads();

    // ---------------- write m out: b128 from LDS -> b128 to global ----------------
    #pragma unroll 1
    for (int i = 0; i < 16; ++i) {
        const v4f* __restrict__ src = (const v4f*)&mt[i * LDSS];          // 16B aligned
        v4f* __restrict__ dst = (v4f*)(out_m + (size_t)(b0 + i) * NMOD);
        dst[lane]      = src[lane];
        dst[lane + 32] = src[lane + 32];
    }
    __syncthreads();

    // ---------------- c = m @ cmMat^T : [16 x 256] x [256 x 70->80] ----------------
    #pragma unroll 1
    for (int nt = 0; nt < 5; ++nt) {
        const int  n  = nt * 16 + l15;
        const bool nv = (n < NCMP);
        const float* __restrict__ cmr = cm + (size_t)(nv ? n : 0) * NMOD + 2 * half;
        const float* __restrict__ ar  = &mt[l15 * LDSS + 2 * half];

        v8f acc = {};
        #pragma unroll 16
        for (int kc = 0; kc < 64; ++kc) {
            const v2f a = *(const v2f*)(ar + kc * 4);            // ds_load_b64, imm offset
            v2f bb;
            if (nv) {
                bb = *(const v2f*)(cmr + kc * 4);                // L2-resident cmMat
            } else {
                bb[0] = 0.f; bb[1] = 0.f;
            }
            acc = __builtin_amdgcn_wmma_f32_16x16x4_f32(
                      false, a, false, bb, (short)0, acc, false, false);
        }

        if (nv) {
            #pragma unroll
            for (int r = 0; r < 8; ++r)
                out_c[(size_t)(b0 + r + 8 * half) * NCMP + n] = acc[r];
        }
    }
}

extern "C" void kernel_launch(void* const* d_in, const int* in_sizes, int n_in,
                              void* d_out, int out_size, void* d_ws, size_t ws_size,
                              hipStream_t stream) {
    (void)in_sizes; (void)n_in; (void)d_ws; (void)ws_size; (void)out_size;
    const float* x   = (const float*)d_in[0];
    const float* W1  = (const float*)d_in[1];
    const float* W2  = (const float*)d_in[2];
    const float* b2  = (const float*)d_in[3];
    const float* cm  = (const float*)d_in[4];
    float* out_m = (float*)d_out;                        // [B, M]
    float* out_c = out_m + (size_t)NB * NMOD;            // [B, C]

    dim3 grid(NB / 16);   // 1024 tiles of 16 cells
    dim3 block(32);       // one wave32 per tile
    flux_fused_kernel<<<grid, block, 0, stream>>>(x, W1, W2, b2, cm, out_m, out_c);
}